// Net_57724360458327
// MI455X (gfx1250) — compile-verified
//
#include <hip/hip_runtime.h>
#include <hip/hip_bf16.h>
#include <math.h>

// Problem sizes (compile-time constants of the reference).
#define NN1 10000
#define NN2 3000
#define NN3 800
#define NN4 200
#define EE1 80000
#define EE2 24000
#define EE3 6400
#define EE4 1600

typedef __attribute__((ext_vector_type(2))) float v2f;
typedef __attribute__((ext_vector_type(8))) float v8f;

static inline int cdiv(int a, int b) { return (a + b - 1) / b; }

__device__ __forceinline__ float eluf(float x) { return x > 0.f ? x : expm1f(x); }

// ---------------------------------------------------------------- fill
__global__ void fill_kernel(float* __restrict__ p, float v, int n) {
  int i = blockIdx.x * blockDim.x + threadIdx.x;
  if (i < n) p[i] = v;
}

// ---------------------------------------------------------------- spline edge pass
// One 32-lane slot per edge; lane = output channel. Source-feature row is
// cached in registers (CIN is a template constant), the 8-corner x CIN FMA
// loop fully unrolls; W reads are lane-coalesced (stride-32 rows).
template <int CIN>
__global__ void spline_edge_kernel(const float* __restrict__ x,
                                   const int* __restrict__ ei,       // (2,E): row0=dst,row1=src
                                   const float* __restrict__ pseudo, // (E,3)
                                   const float* __restrict__ W,      // (125,CIN,32)
                                   float* __restrict__ agg,          // (N,32)
                                   float* __restrict__ deg,          // (N)
                                   int E) {
  int t = blockIdx.x * blockDim.x + threadIdx.x;
  int e = t >> 5;
  int o = t & 31;
  if (e >= E) return;
  int dst = ei[e];
  int src = ei[E + e];

  float f0, f1, f2;
  int g0, g1, g2;
  {
    float v = pseudo[e * 3 + 0] * 4.f;
    float gf = fminf(fmaxf(floorf(v), 0.f), 3.f);
    g0 = (int)gf; f0 = v - gf;
    v = pseudo[e * 3 + 1] * 4.f;
    gf = fminf(fmaxf(floorf(v), 0.f), 3.f);
    g1 = (int)gf; f1 = v - gf;
    v = pseudo[e * 3 + 2] * 4.f;
    gf = fminf(fmaxf(floorf(v), 0.f), 3.f);
    g2 = (int)gf; f2 = v - gf;
  }

  // cache source feature row in registers
  const float* xs = x + (size_t)src * CIN;
  float xr[CIN];
#pragma unroll
  for (int i = 0; i < CIN; ++i) xr[i] = xs[i];

  float acc = 0.f;
#pragma unroll
  for (int s = 0; s < 8; ++s) {
    int b0 = s & 1, b1 = (s >> 1) & 1, b2 = (s >> 2) & 1;
    float bw = (b0 ? f0 : 1.f - f0) * (b1 ? f1 : 1.f - f1) * (b2 ? f2 : 1.f - f2);
    int idx = (g0 + b0) + (g1 + b1) * 5 + (g2 + b2) * 25;
    const float* Wk = W + ((size_t)idx * CIN) * 32 + o;
    float d = 0.f;
#pragma unroll
    for (int i = 0; i < CIN; ++i) d = fmaf(xr[i], Wk[(size_t)i * 32], d);
    acc = fmaf(bw, d, acc);
  }
  atomicAdd(&agg[(size_t)dst * 32 + o], acc);
  if (o == 0) atomicAdd(&deg[dst], 1.f);
}

// ---------------------------------------------------------------- WMMA f32 GEMM (+fused combine)
// out[m, colOff+c] = ACT( A[m,:K] @ Wm[:,c] + bias[c] + (agg[m,c]/max(deg[m],1) if HAS_AGG) )
// One wave per 16-row tile; V_WMMA_F32_16X16X4_F32, fully unrolled K chain.
// A-frag (16x4 f32, ISA 7.12.2): lane l -> row l%16; lane-half picks K pair {0,1}/{2,3}.
// B-frag (4x16): same half-split K convention; col = lane%16.
// D/C (16x16 f32, 8 VGPRs): vgpr r, lane l -> row r+8*(l/16), col l%16.
// All loads use clamped (always-valid) addresses + value selects, so no
// exec-mask branches occur before/inside the WMMA chain (EXEC stays all-1s).
template <int K, int NC, bool HAS_AGG, bool ACT>
__global__ __launch_bounds__(32) void wmma_gemm_kernel(
    const float* __restrict__ A,     // (M,K)
    const float* __restrict__ Wm,    // (K,NC)
    const float* __restrict__ bias,  // (NC)
    const float* __restrict__ agg,   // (M,32)
    const float* __restrict__ deg,   // (M)
    float* __restrict__ out, int M, int ldOut, int colOff) {
  int l = threadIdx.x;
  int half = l >> 4;
  int lm = l & 15;
  int mbase = blockIdx.x * 16;
  int row = mbase + lm;
  int rowc = row < M ? row : (M - 1);  // clamped address; store is masked

  constexpr int NTILE = (NC > 16) ? 2 : 1;
  const int cw = (NC < 16) ? (lm & (NC - 1)) : lm;  // clamped B column (NC power of two)
  const bool cvalid = (NC >= 16) || (lm < NC);

  v8f d0 = {0.f, 0.f, 0.f, 0.f, 0.f, 0.f, 0.f, 0.f};
  v8f d1 = {0.f, 0.f, 0.f, 0.f, 0.f, 0.f, 0.f, 0.f};

#pragma unroll
  for (int k = 0; k < K; k += 4) {
    int kb = k + half * 2;
    v2f a;
    a.x = A[(size_t)rowc * K + kb];
    a.y = A[(size_t)rowc * K + kb + 1];

    v2f b0;
    float w0 = Wm[(size_t)(kb) * NC + cw];
    float w1 = Wm[(size_t)(kb + 1) * NC + cw];
    b0.x = cvalid ? w0 : 0.f;
    b0.y = cvalid ? w1 : 0.f;
    d0 = __builtin_amdgcn_wmma_f32_16x16x4_f32(false, a, false, b0, (short)0, d0, false, false);

    if (NTILE == 2) {
      v2f b1;
      b1.x = Wm[(size_t)(kb) * NC + 16 + lm];
      b1.y = Wm[(size_t)(kb + 1) * NC + 16 + lm];
      d1 = __builtin_amdgcn_wmma_f32_16x16x4_f32(false, a, false, b1, (short)0, d1, false, false);
    }
  }

#pragma unroll
  for (int r = 0; r < 8; ++r) {
    int m = mbase + r + 8 * half;
    if (m >= M) continue;
    float degv = 1.f;
    if (HAS_AGG) degv = fmaxf(deg[m], 1.f);
    if (cvalid) {
      int c = lm;
      float v = d0[r] + bias[cw];
      if (HAS_AGG) v += agg[(size_t)m * 32 + c] / degv;
      if (ACT) v = eluf(v);
      out[(size_t)m * ldOut + colOff + c] = v;
    }
    if (NTILE == 2) {
      int c = 16 + lm;
      float v = d1[r] + bias[c];
      if (HAS_AGG) v += agg[(size_t)m * 32 + c] / degv;
      if (ACT) v = eluf(v);
      out[(size_t)m * ldOut + colOff + c] = v;
    }
  }
}

// ---------------------------------------------------------------- scalar combine (Cin==1 conv1)
__global__ void combine_scalar_kernel(const float* __restrict__ x,
                                      const float* __restrict__ root,  // (1,32)
                                      const float* __restrict__ bias,
                                      const float* __restrict__ agg,
                                      const float* __restrict__ deg,
                                      float* __restrict__ out, int N) {
  int t = blockIdx.x * blockDim.x + threadIdx.x;
  if (t >= N * 32) return;
  int n = t >> 5, o = t & 31;
  float v = agg[t] / fmaxf(deg[n], 1.f) + bias[o];
  v = fmaf(x[n], root[o], v);
  out[t] = eluf(v);
}

// ---------------------------------------------------------------- segment max (voxel pool)
__device__ __forceinline__ void atomicMaxFloat(float* addr, float val) {
  unsigned int* ua = reinterpret_cast<unsigned int*>(addr);
  unsigned int old = *ua;
  while (__uint_as_float(old) < val) {
    unsigned int assumed = old;
    old = atomicCAS(ua, assumed, __float_as_uint(val));
    if (old == assumed) break;
  }
}

__global__ void seg_max_kernel(const float* __restrict__ x, const int* __restrict__ cl,
                               float* __restrict__ out, int Nfine) {
  int t = blockIdx.x * blockDim.x + threadIdx.x;
  if (t >= Nfine * 32) return;
  int n = t >> 5, o = t & 31;
  atomicMaxFloat(&out[(size_t)cl[n] * 32 + o], x[t]);
}

// ---------------------------------------------------------------- upsample gather (concat lo half)
__global__ void gather_kernel(const float* __restrict__ src, const int* __restrict__ cl,
                              float* __restrict__ u, int Nfine) {
  int t = blockIdx.x * blockDim.x + threadIdx.x;
  if (t >= Nfine * 32) return;
  int n = t >> 5, o = t & 31;
  u[(size_t)n * 40 + o] = src[(size_t)cl[n] * 32 + o];
}

// ---------------------------------------------------------------- head: fc2 (32->3) + log_softmax
__global__ void head_kernel(const float* __restrict__ x, const float* __restrict__ w,  // (32,3)
                            const float* __restrict__ b, float* __restrict__ out, int N) {
  int n = blockIdx.x * blockDim.x + threadIdx.x;
  if (n >= N) return;
  const float* r = x + (size_t)n * 32;
  float l0 = b[0], l1 = b[1], l2 = b[2];
#pragma unroll
  for (int i = 0; i < 32; ++i) {
    float xi = r[i];
    l0 = fmaf(xi, w[i * 3 + 0], l0);
    l1 = fmaf(xi, w[i * 3 + 1], l1);
    l2 = fmaf(xi, w[i * 3 + 2], l2);
  }
  float m = fmaxf(l0, fmaxf(l1, l2));
  float lse = m + logf(expf(l0 - m) + expf(l1 - m) + expf(l2 - m));
  out[(size_t)n * 3 + 0] = l0 - lse;
  out[(size_t)n * 3 + 1] = l1 - lse;
  out[(size_t)n * 3 + 2] = l2 - lse;
}

// ---------------------------------------------------------------- host-side conv helper
template <int CIN>
static void spline_conv(const float* xin, const int* ei, const float* pseudo, int E,
                        const float* W, const float* root, const float* bias, float* agg,
                        float* deg, float* out, int N, hipStream_t s) {
  fill_kernel<<<cdiv(N * 32, 256), 256, 0, s>>>(agg, 0.f, N * 32);
  fill_kernel<<<cdiv(N, 256), 256, 0, s>>>(deg, 0.f, N);
  spline_edge_kernel<CIN><<<cdiv(E * 32, 256), 256, 0, s>>>(xin, ei, pseudo, W, agg, deg, E);
  if (CIN == 1) {
    combine_scalar_kernel<<<cdiv(N * 32, 256), 256, 0, s>>>(xin, root, bias, agg, deg, out, N);
  } else {
    constexpr int KK = (CIN == 1) ? 4 : CIN;  // avoid instantiating K=1 path
    wmma_gemm_kernel<KK, 32, true, true>
        <<<cdiv(N, 16), 32, 0, s>>>(xin, root, bias, agg, deg, out, N, 32, 0);
  }
}

extern "C" void kernel_launch(void* const* d_in, const int* in_sizes, int n_in, void* d_out,
                              int out_size, void* d_ws, size_t ws_size, hipStream_t stream) {
  (void)in_sizes; (void)n_in; (void)out_size; (void)ws_size;
  // setup_inputs() dict order:
  const float* x   = (const float*)d_in[0];
  const int*   ei1 = (const int*)d_in[1];
  const float* p1  = (const float*)d_in[2];
  const int*   ei2 = (const int*)d_in[3];
  const float* p2  = (const float*)d_in[4];
  const int*   ei3 = (const int*)d_in[5];
  const float* p3  = (const float*)d_in[6];
  const int*   ei4 = (const int*)d_in[7];
  const float* p4  = (const float*)d_in[8];
  const int*   cl1 = (const int*)d_in[9];
  const int*   cl2 = (const int*)d_in[10];
  const int*   cl3 = (const int*)d_in[11];
  // d_in[12] = n4 scalar (==NN4, compile-time)
  const float *W1 = (const float*)d_in[13], *root1 = (const float*)d_in[14], *b1 = (const float*)d_in[15];
  const float *W2 = (const float*)d_in[16], *root2 = (const float*)d_in[17], *b2 = (const float*)d_in[18];
  const float *W3 = (const float*)d_in[19], *root3 = (const float*)d_in[20], *b3 = (const float*)d_in[21];
  const float *W42 = (const float*)d_in[22], *root42 = (const float*)d_in[23], *b42 = (const float*)d_in[24];
  const float *W5 = (const float*)d_in[25], *root5 = (const float*)d_in[26], *b5 = (const float*)d_in[27];
  const float *W6 = (const float*)d_in[28], *root6 = (const float*)d_in[29], *b6 = (const float*)d_in[30];
  const float *W7 = (const float*)d_in[31], *root7 = (const float*)d_in[32], *b7 = (const float*)d_in[33];
  const float *fc1_w = (const float*)d_in[34], *fc1_b = (const float*)d_in[35];
  const float *fc2_w = (const float*)d_in[36], *fc2_b = (const float*)d_in[37];
  const float *s1_w = (const float*)d_in[38], *s1_b = (const float*)d_in[39];
  const float *s2_w = (const float*)d_in[40], *s2_b = (const float*)d_in[41];
  const float *s3_w = (const float*)d_in[42], *s3_b = (const float*)d_in[43];

  float* ws = (float*)d_ws;
  size_t off = 0;
  auto carve = [&](size_t n) { float* p = ws + off; off += n; return p; };
  float* x1  = carve((size_t)NN1 * 32);
  float* x2p = carve((size_t)NN2 * 32);
  float* x2  = carve((size_t)NN2 * 32);
  float* x3p = carve((size_t)NN3 * 32);
  float* x3  = carve((size_t)NN3 * 32);
  float* x4p = carve((size_t)NN4 * 32);
  float* x4c = carve((size_t)NN4 * 32);
  float* x4f = carve((size_t)NN4 * 32);
  float* u3  = carve((size_t)NN3 * 40);
  float* x3d = carve((size_t)NN3 * 32);
  float* u2  = carve((size_t)NN2 * 40);
  float* x2d = carve((size_t)NN2 * 32);
  float* u1  = carve((size_t)NN1 * 40);
  float* x1d = carve((size_t)NN1 * 32);
  float* agg = carve((size_t)NN1 * 32);
  float* deg = carve((size_t)NN1);

  hipStream_t s = stream;

  // ---- encoder
  spline_conv<1>(x, ei1, p1, EE1, W1, root1, b1, agg, deg, x1, NN1, s);

  fill_kernel<<<cdiv(NN2 * 32, 256), 256, 0, s>>>(x2p, -1e30f, NN2 * 32);
  seg_max_kernel<<<cdiv(NN1 * 32, 256), 256, 0, s>>>(x1, cl1, x2p, NN1);
  spline_conv<32>(x2p, ei2, p2, EE2, W2, root2, b2, agg, deg, x2, NN2, s);

  fill_kernel<<<cdiv(NN3 * 32, 256), 256, 0, s>>>(x3p, -1e30f, NN3 * 32);
  seg_max_kernel<<<cdiv(NN2 * 32, 256), 256, 0, s>>>(x2, cl2, x3p, NN2);
  spline_conv<32>(x3p, ei3, p3, EE3, W3, root3, b3, agg, deg, x3, NN3, s);

  fill_kernel<<<cdiv(NN4 * 32, 256), 256, 0, s>>>(x4p, -1e30f, NN4 * 32);
  seg_max_kernel<<<cdiv(NN3 * 32, 256), 256, 0, s>>>(x3, cl3, x4p, NN3);
  spline_conv<32>(x4p, ei4, p4, EE4, W42, root42, b42, agg, deg, x4c, NN4, s);

  // fc1 + ELU
  wmma_gemm_kernel<32, 32, false, true>
      <<<cdiv(NN4, 16), 32, 0, s>>>(x4c, fc1_w, fc1_b, nullptr, nullptr, x4f, NN4, 32, 0);

  // ---- decoder
  gather_kernel<<<cdiv(NN3 * 32, 256), 256, 0, s>>>(x4f, cl3, u3, NN3);
  wmma_gemm_kernel<32, 8, false, false>
      <<<cdiv(NN3, 16), 32, 0, s>>>(x3, s3_w, s3_b, nullptr, nullptr, u3, NN3, 40, 32);
  spline_conv<40>(u3, ei3, p3, EE3, W5, root5, b5, agg, deg, x3d, NN3, s);

  gather_kernel<<<cdiv(NN2 * 32, 256), 256, 0, s>>>(x3d, cl2, u2, NN2);
  wmma_gemm_kernel<32, 8, false, false>
      <<<cdiv(NN2, 16), 32, 0, s>>>(x2, s2_w, s2_b, nullptr, nullptr, u2, NN2, 40, 32);
  spline_conv<40>(u2, ei2, p2, EE2, W6, root6, b6, agg, deg, x2d, NN2, s);

  gather_kernel<<<cdiv(NN1 * 32, 256), 256, 0, s>>>(x2d, cl1, u1, NN1);
  wmma_gemm_kernel<32, 8, false, false>
      <<<cdiv(NN1, 16), 32, 0, s>>>(x1, s1_w, s1_b, nullptr, nullptr, u1, NN1, 40, 32);
  spline_conv<40>(u1, ei1, p1, EE1, W7, root7, b7, agg, deg, x1d, NN1, s);

  // ---- head
  head_kernel<<<cdiv(NN1, 256), 256, 0, s>>>(x1d, fc2_w, fc2_b, (float*)d_out, NN1);
}